// AttentionBlock_35029753266787
// MI455X (gfx1250) — compile-verified
//
#include <hip/hip_runtime.h>
#include <math.h>

typedef __attribute__((ext_vector_type(2))) float v2f;
typedef __attribute__((ext_vector_type(4))) float v4f;
typedef __attribute__((ext_vector_type(8))) float v8f;

#define BATCH 64
#define CCH 3
#define IMG 512
#define PATCH_DIM 768   // 16*16*3
#define IN_F 256
#define HD 512
#define EPSV 1e-5f

// ---------------------------------------------------------------------------
// Kernel 1: strided patch-position reduction (the memory-bound hot spot:
// 201 MB of feature read once -> ~8.6 us at 23.3 TB/s).
// psum[b][(p1*16+p2)*3+c] = sum over 32x32 grid of feature[b][c][16h+p1][16w+p2]
// One block per (b, c, p1): reads 32 rows x 512 cols = 64KB.
//  - b128 loads, fully coalesced (512B/wave/instr), NT hint (feature > 192MB L2)
//  - 3-step shfl_xor butterfly folds 8 lanes before LDS -> 8x fewer ds atomics
// ---------------------------------------------------------------------------
__global__ void k_patch_reduce(const float* __restrict__ feature,
                               float* __restrict__ psum) {
    int blk = blockIdx.x;               // b*48 + c*16 + p1
    int p1  = blk & 15;
    int c   = (blk >> 4) % 3;
    int b   = blk / 48;
    const float* plane = feature + ((size_t)(b * CCH + c)) * IMG * IMG;

    int t     = threadIdx.x;            // 256 threads
    int col4  = t & 127;                // float4 group 0..127
    int x     = col4 * 4;
    int rhalf = t >> 7;                 // 0..1

    float a0 = 0.f, a1 = 0.f, a2 = 0.f, a3 = 0.f;
    #pragma unroll
    for (int i = 0; i < 16; ++i) {
        int y = p1 + 16 * (rhalf + 2 * i);     // covers all 32 rows of class p1
        const v4f v = __builtin_nontemporal_load(
            (const v4f*)(plane + (size_t)y * IMG + x));
        a0 += v.x; a1 += v.y; a2 += v.z; a3 += v.w;
    }

    // Butterfly over lanes differing in bits 2..4: the 8 lanes of a wave with
    // equal (lane % 4) all target the same 4 p2-bins.
    #pragma unroll
    for (int m = 4; m <= 16; m <<= 1) {
        a0 += __shfl_xor(a0, m, 32);
        a1 += __shfl_xor(a1, m, 32);
        a2 += __shfl_xor(a2, m, 32);
        a3 += __shfl_xor(a3, m, 32);
    }

    __shared__ float bins[16];
    if (t < 16) bins[t] = 0.f;
    __syncthreads();
    int lane = t & 31;
    if (lane < 4) {                     // one representative per 8-lane group
        int p2b = lane * 4;             // == ((t % 4) * 4) for lanes 0..3
        atomicAdd(&bins[p2b + 0], a0);
        atomicAdd(&bins[p2b + 1], a1);
        atomicAdd(&bins[p2b + 2], a2);
        atomicAdd(&bins[p2b + 3], a3);
    }
    __syncthreads();
    if (t < 16) {
        int k = (p1 * 16 + t) * 3 + c;  // unique owner per (b,k) -> plain store
        psum[b * PATCH_DIM + k] = bins[t];
    }
}

// ---------------------------------------------------------------------------
// fp32 WMMA helper tile: one wave computes one 16x16 f32 tile of A[MxK]@B[KxN]
// using V_WMMA_F32_16X16X4_F32, K stepped by 4.
//   A layout: lanes 0-15 hold M=lane {K, K+1}; lanes 16-31 hold M=lane-16 {K+2, K+3}
//   B layout: vgpr0 lanes0-15 = row K (N=lane), lanes16-31 = row K+2; vgpr1 = K+1/K+3
//   C/D: vgpr j -> row j (lanes 0-15) / row j+8 (lanes 16-31), col = lane%16
// ---------------------------------------------------------------------------
__device__ __forceinline__ v8f wmma_tile_f32(const float* __restrict__ A, int lda,
                                             const float* __restrict__ B, int ldb,
                                             int K, int lane) {
    int mrow  = lane & 15;
    int khalf = (lane >> 4) << 1;               // 0 or 2
    const float* Ab = A + (size_t)mrow * lda + khalf;
    const float* Bb = B + (lane & 15);
    v8f acc = {};
    for (int k = 0; k < K; k += 4) {
        v2f a;  a.x  = Ab[k];                  a.y  = Ab[k + 1];
        v2f bb; bb.x = Bb[(size_t)(k + khalf) * ldb];
                bb.y = Bb[(size_t)(k + khalf + 1) * ldb];
        acc = __builtin_amdgcn_wmma_f32_16x16x4_f32(false, a, false, bb,
                                                    (short)0, acc, false, false);
    }
    return acc;
}

// ---------------------------------------------------------------------------
// Kernel 2: s[64,256] = psum[64,768] @ W_patch[768,256] + 1024*b_patch
// grid 64 = 4 mtiles * 16 ntiles, block = 1 wave
// ---------------------------------------------------------------------------
__global__ void k_gemm1(const float* __restrict__ psum,
                        const float* __restrict__ W_patch,
                        const float* __restrict__ b_patch,
                        float* __restrict__ s) {
    int mt = blockIdx.x >> 4, nt = blockIdx.x & 15;
    int lane = threadIdx.x;
    v8f acc = wmma_tile_f32(psum + (size_t)mt * 16 * PATCH_DIM, PATCH_DIM,
                            W_patch + nt * 16, IN_F, PATCH_DIM, lane);
    int rbase = mt * 16 + ((lane >> 4) << 3);
    int col   = nt * 16 + (lane & 15);
    float bias = 1024.0f * b_patch[col];
    #pragma unroll
    for (int j = 0; j < 8; ++j)
        s[(size_t)(rbase + j) * IN_F + col] = acc[j] + bias;
}

// ---------------------------------------------------------------------------
// Kernel 3: t[64,512] = s[64,256] @ W[256,512]
// grid 128 = 4 mtiles * 32 ntiles, block = 1 wave
// ---------------------------------------------------------------------------
__global__ void k_gemm2(const float* __restrict__ s,
                        const float* __restrict__ W,
                        float* __restrict__ tbuf) {
    int mt = blockIdx.x >> 5, nt = blockIdx.x & 31;
    int lane = threadIdx.x;
    v8f acc = wmma_tile_f32(s + (size_t)mt * 16 * IN_F, IN_F,
                            W + nt * 16, HD, IN_F, lane);
    int rbase = mt * 16 + ((lane >> 4) << 3);
    int col   = nt * 16 + (lane & 15);
    #pragma unroll
    for (int j = 0; j < 8; ++j)
        tbuf[(size_t)(rbase + j) * HD + col] = acc[j];
}

// ---------------------------------------------------------------------------
// Kernel 4: u = t @ W_out [64,512]x[512,256]; x = 0.25*(u+b_out)+0.75*token;
// LayerNorm(x)*gamma+beta -> out[64,256].
// grid 4 (one per 16-row mtile), block 256 = 8 waves, each wave 2 ntiles.
// ---------------------------------------------------------------------------
__global__ void k_gemm3_ln(const float* __restrict__ tbuf,
                           const float* __restrict__ W_out,
                           const float* __restrict__ b_out,
                           const float* __restrict__ token,
                           const float* __restrict__ gamma,
                           const float* __restrict__ beta,
                           float* __restrict__ out) {
    __shared__ float xbuf[16 * IN_F];
    __shared__ float pm[16][17];
    __shared__ float pq[16][17];
    __shared__ float mu_s[16], rs_s[16];

    int mt   = blockIdx.x;
    int tid  = threadIdx.x;
    int wave = tid >> 5, lane = tid & 31;

    #pragma unroll
    for (int w = 0; w < 2; ++w) {
        int nt = wave * 2 + w;
        v8f acc = wmma_tile_f32(tbuf + (size_t)mt * 16 * HD, HD,
                                W_out + nt * 16, IN_F, HD, lane);
        int rbase = (lane >> 4) << 3;
        int col   = nt * 16 + (lane & 15);
        #pragma unroll
        for (int j = 0; j < 8; ++j) {
            int r    = rbase + j;
            int grow = mt * 16 + r;
            float x  = 0.25f * (acc[j] + b_out[col])
                     + 0.75f * token[(size_t)grow * IN_F + col];
            xbuf[r * IN_F + col] = x;
        }
    }
    __syncthreads();

    // LayerNorm over IN_F=256 per row; 16 threads per row, 16 cols each
    int r = tid >> 4, sub = tid & 15;
    float sm = 0.f, sq = 0.f;
    #pragma unroll
    for (int j = 0; j < 16; ++j) {
        float v = xbuf[r * IN_F + sub * 16 + j];
        sm += v; sq += v * v;
    }
    pm[r][sub] = sm; pq[r][sub] = sq;
    __syncthreads();
    if (sub == 0) {
        float m = 0.f, q = 0.f;
        #pragma unroll
        for (int j = 0; j < 16; ++j) { m += pm[r][j]; q += pq[r][j]; }
        m /= (float)IN_F;
        q  = q / (float)IN_F - m * m;
        mu_s[r] = m;
        rs_s[r] = rsqrtf(q + EPSV);
    }
    __syncthreads();
    float m = mu_s[r], rs = rs_s[r];
    #pragma unroll
    for (int j = 0; j < 16; ++j) {
        int col = sub * 16 + j;
        float v = xbuf[r * IN_F + col];
        out[(size_t)(mt * 16 + r) * IN_F + col] = (v - m) * rs * gamma[col] + beta[col];
    }
}

// ---------------------------------------------------------------------------
extern "C" void kernel_launch(void* const* d_in, const int* in_sizes, int n_in,
                              void* d_out, int out_size, void* d_ws, size_t ws_size,
                              hipStream_t stream) {
    (void)in_sizes; (void)n_in; (void)out_size; (void)ws_size;
    const float* token   = (const float*)d_in[0];
    const float* feature = (const float*)d_in[1];
    const float* W_patch = (const float*)d_in[2];
    const float* b_patch = (const float*)d_in[3];
    const float* W       = (const float*)d_in[4];
    // d_in[5] = a_self, d_in[6] = a_neigh : mathematically dead (softmax over size-1 axis)
    const float* W_out   = (const float*)d_in[7];
    const float* b_out   = (const float*)d_in[8];
    const float* gamma   = (const float*)d_in[9];
    const float* beta    = (const float*)d_in[10];
    float* out = (float*)d_out;

    float* psum = (float*)d_ws;                 // 64*768
    float* sbuf = psum + BATCH * PATCH_DIM;     // 64*256
    float* tbuf = sbuf + BATCH * IN_F;          // 64*512

    k_patch_reduce<<<dim3(BATCH * CCH * 16), dim3(256), 0, stream>>>(feature, psum);
    k_gemm1<<<dim3(64), dim3(32), 0, stream>>>(psum, W_patch, b_patch, sbuf);
    k_gemm2<<<dim3(128), dim3(32), 0, stream>>>(sbuf, W, tbuf);
    k_gemm3_ln<<<dim3(4), dim3(256), 0, stream>>>(tbuf, W_out, b_out, token,
                                                  gamma, beta, out);
}